// CRF_936302870858
// MI455X (gfx1250) — compile-verified
//
#include <hip/hip_runtime.h>
#include <hip/hip_bf16.h>

#ifndef __has_builtin
#define __has_builtin(x) 0
#endif

// ---- CDNA5 async global->LDS staging (gfx1250) --------------------------
#if __has_builtin(__builtin_amdgcn_global_load_async_to_lds_b64)
#define HAVE_ASYNC_LDS 1
#else
#define HAVE_ASYNC_LDS 0
#endif

#if __has_builtin(__builtin_amdgcn_s_wait_asynccnt)
#define WAIT_ASYNC(n) __builtin_amdgcn_s_wait_asynccnt(n)
#else
#define WAIT_ASYNC(n) asm volatile("s_wait_asynccnt " #n ::: "memory")
#endif

// ---- CDNA5 WMMA (f32 16x16x4) for the final reduction -------------------
#if __has_builtin(__builtin_amdgcn_wmma_f32_16x16x4_f32)
#define HAVE_WMMA_F32 1
#else
#define HAVE_WMMA_F32 0
#endif

namespace {
constexpr int Tn = 2048;
constexpr int Bn = 256;
constexpr int Mn = 32;
constexpr float LOG2E = 1.44269504088896340736f;
constexpr float LN2   = 0.693147180559945309417f;
}

#if HAVE_ASYNC_LDS
// b64 async copy: payload type is a 2x i32 vector per the builtin's signature
// (param 1: AS(1) v2i*, param 2: AS(3) v2i*, then imm offset + cpol).
__device__ __forceinline__ void async_copy_b64(const void* gsrc, void* ldst) {
  typedef int v2i __attribute__((vector_size(8)));
  typedef __attribute__((address_space(1))) v2i* gp_t;
  typedef __attribute__((address_space(3))) v2i* lp_t;
  __builtin_amdgcn_global_load_async_to_lds_b64((gp_t)gsrc, (lp_t)ldst, 0, 0);
}
#endif

// =========================================================================
// Kernel A: log-semiring forward scan. One wave32 per pair of batch rows.
// lane j = state j. alpha'[j] = em[j] + M + ln( sum_i exp2( (alpha_i-M)*L
//            + (trans_ij*L)*rw ) ), M = max_i alpha + tmax*rw  (args <= 0).
// =========================================================================
__global__ __launch_bounds__(32) void crf_forward_kernel(
    const float* __restrict__ em, const float* __restrict__ wgt,
    const int* __restrict__ msk, const float* __restrict__ trans,
    const float* __restrict__ st, const float* __restrict__ et,
    float* __restrict__ logZ)
{
  const int j  = threadIdx.x;      // state index (wave32 lane)
  const int b0 = blockIdx.x * 2;   // two batch rows per wave (dual-stream ILP)
  const int b1 = b0 + 1;

  // Per-lane: column j of transitions, pre-scaled by log2(e) for exp2 domain.
  float tcl[Mn];
  float tmax = -INFINITY;
#pragma unroll
  for (int i = 0; i < Mn; ++i) {
    float v = trans[i * Mn + j];
    tmax = fmaxf(tmax, v);
    tcl[i] = v * LOG2E;
  }
#pragma unroll
  for (int off = 16; off; off >>= 1)
    tmax = fmaxf(tmax, __shfl_xor(tmax, off, 32));

  float alpha0 = st[j] + em[(0 * Bn + b0) * Mn + j];
  float alpha1 = st[j] + em[(0 * Bn + b1) * Mn + j];

  // One forward step for one stream.
  auto step = [&](float alpha, float emt, float rw) -> float {
    float am = alpha;
#pragma unroll
    for (int off = 16; off; off >>= 1)
      am = fmaxf(am, __shfl_xor(am, off, 32));
    const float Mt  = __builtin_fmaf(tmax, rw, am);  // stability shift
    const float nML = -Mt * LOG2E;
    float a0 = 0.f, a1 = 0.f, a2 = 0.f, a3 = 0.f;
#pragma unroll
    for (int i = 0; i < Mn; i += 4) {
      float s0 = __builtin_fmaf(__shfl(alpha, i + 0, 32), LOG2E, nML);
      float s1 = __builtin_fmaf(__shfl(alpha, i + 1, 32), LOG2E, nML);
      float s2 = __builtin_fmaf(__shfl(alpha, i + 2, 32), LOG2E, nML);
      float s3 = __builtin_fmaf(__shfl(alpha, i + 3, 32), LOG2E, nML);
      a0 += __builtin_amdgcn_exp2f(__builtin_fmaf(tcl[i + 0], rw, s0));
      a1 += __builtin_amdgcn_exp2f(__builtin_fmaf(tcl[i + 1], rw, s1));
      a2 += __builtin_amdgcn_exp2f(__builtin_fmaf(tcl[i + 2], rw, s2));
      a3 += __builtin_amdgcn_exp2f(__builtin_fmaf(tcl[i + 3], rw, s3));
    }
    float s = (a0 + a1) + (a2 + a3);
    return emt + __builtin_fmaf(__builtin_amdgcn_logf(s), LN2, Mt);
  };

  auto do_t = [&](float emt0, float emt1, float w0, float w1, int m0, int m1) {
    if (m0) alpha0 = step(alpha0, emt0, 1.0f / w0);  // masked step = identity
    if (m1) alpha1 = step(alpha1, emt1, 1.0f / w1);
  };

#if HAVE_ASYNC_LDS
  // 8-deep LDS ring: slot = [64 em floats (b0,b1)][w(t-1) b0,b1][mask(t) b0,b1]
  constexpr int DEPTH = 8;
  constexpr int RS    = 68;
  __shared__ float ring[DEPTH * RS];

  auto issue = [&](int tp) {  // stage data needed by iteration tp
    int slot = (tp - 1) & (DEPTH - 1);
    float* row = &ring[slot * RS];
    const float* g = &em[(tp * Bn + b0) * Mn];  // 256B = rows b0 and b1
    async_copy_b64(g + 2 * j, row + 2 * j);
    if (j == 0) {
      async_copy_b64(&wgt[(tp - 1) * Bn + b0], row + 64);
      async_copy_b64(&msk[tp * Bn + b0], row + 66);
    }
  };

#pragma unroll
  for (int tp = 1; tp <= DEPTH; ++tp) issue(tp);

  // Steady state: 3 async ops per group; keep DEPTH-1 groups in flight.
  for (int t = 1; t <= Tn - 1 - DEPTH; ++t) {
    WAIT_ASYNC(21);                       // 3 * (DEPTH-1): group t complete
    asm volatile("" ::: "memory");
    const int slot = (t - 1) & (DEPTH - 1);
    const float* row = &ring[slot * RS];
    float emt0 = row[j], emt1 = row[32 + j];
    float w0 = row[64], w1 = row[65];
    int m0 = __float_as_int(row[66]), m1 = __float_as_int(row[67]);
    do_t(emt0, emt1, w0, w1, m0, m1);
    issue(t + DEPTH);
  }
  // Tail: fewer groups outstanding -> wait for all.
  for (int t = Tn - DEPTH; t < Tn; ++t) {
    WAIT_ASYNC(0);
    asm volatile("" ::: "memory");
    const int slot = (t - 1) & (DEPTH - 1);
    const float* row = &ring[slot * RS];
    float emt0 = row[j], emt1 = row[32 + j];
    float w0 = row[64], w1 = row[65];
    int m0 = __float_as_int(row[66]), m1 = __float_as_int(row[67]);
    do_t(emt0, emt1, w0, w1, m0, m1);
  }
#else
  // Fallback: direct loads with hardware prefetch ahead of the chain.
  for (int t = 1; t < Tn; ++t) {
    if (t + 8 < Tn)
      __builtin_prefetch(&em[((t + 8) * Bn + b0) * Mn + j], 0, 0);
    float emt0 = em[(t * Bn + b0) * Mn + j];
    float emt1 = em[(t * Bn + b1) * Mn + j];
    float w0 = wgt[(t - 1) * Bn + b0], w1 = wgt[(t - 1) * Bn + b1];
    int m0 = msk[t * Bn + b0], m1 = msk[t * Bn + b1];
    do_t(emt0, emt1, w0, w1, m0, m1);
  }
#endif

  // log Z = logsumexp_j(alpha + end_transitions)
  auto lse_out = [&](float alpha) -> float {
    float v = alpha + et[j];
    float vm = v;
#pragma unroll
    for (int off = 16; off; off >>= 1)
      vm = fmaxf(vm, __shfl_xor(vm, off, 32));
    float e = __builtin_amdgcn_exp2f((v - vm) * LOG2E);
#pragma unroll
    for (int off = 16; off; off >>= 1)
      e += __shfl_xor(e, off, 32);
    return __builtin_fmaf(__builtin_amdgcn_logf(e), LN2, vm);
  };
  float z0 = lse_out(alpha0);
  float z1 = lse_out(alpha1);
  if (j == 0) { logZ[b0] = z0; logZ[b1] = z1; }
}

// =========================================================================
// Kernel B: gold-path score. One block per batch element, threads stride t.
// =========================================================================
__global__ __launch_bounds__(256) void crf_score_kernel(
    const float* __restrict__ em, const int* __restrict__ tags,
    const float* __restrict__ wgt, const int* __restrict__ msk,
    const float* __restrict__ trans, const float* __restrict__ st,
    const float* __restrict__ et, float* __restrict__ score)
{
  const int b = blockIdx.x;
  const int tid = threadIdx.x;

  float acc = 0.f;
  int cnt = 0;
  for (int t = tid; t < Tn; t += 256) {
    if (t + 256 < Tn) {
      __builtin_prefetch(&msk[(t + 256) * Bn + b], 0, 0);   // global_prefetch_b8
      __builtin_prefetch(&tags[(t + 256) * Bn + b], 0, 0);
    }
    int m = msk[t * Bn + b];
    cnt += (m != 0) ? 1 : 0;
    float mf = (float)m;
    int tg = (m != 0) ? tags[t * Bn + b] : 1;
    if (t == 0) {
      acc += st[tg] + em[(0 * Bn + b) * Mn + tg] * mf;
    } else {
      int mp  = msk[(t - 1) * Bn + b];
      int tgp = (mp != 0) ? tags[(t - 1) * Bn + b] : 1;
      acc += trans[tgp * Mn + tg] * mf / wgt[(t - 1) * Bn + b]
           + em[(t * Bn + b) * Mn + tg] * mf;
    }
  }

  // deterministic block reduction (float sum + int mask count)
#pragma unroll
  for (int off = 16; off; off >>= 1) {
    acc += __shfl_xor(acc, off, 32);
    cnt += __shfl_xor(cnt, off, 32);
  }
  __shared__ float sacc[8];
  __shared__ int   scnt[8];
  const int wid = tid >> 5, lid = tid & 31;
  if (lid == 0) { sacc[wid] = acc; scnt[wid] = cnt; }
  __syncthreads();
  if (tid == 0) {
    float tot = 0.f;
    int c = 0;
    for (int i = 0; i < 8; ++i) { tot += sacc[i]; c += scnt[i]; }
    const int se = c - 1;                       // seq_ends[b]
    const int ml = msk[se * Bn + b];
    const int lt = (ml != 0) ? tags[se * Bn + b] : 1;
    score[b] = tot + et[lt];
  }
}

// =========================================================================
// Kernel C: final reduction sum_b (logZ[b] - score[b]) on the matrix pipe.
// D = 1^T x B computes column sums: chain 4 x V_WMMA_F32_16X16X4_F32 with an
// all-ones A (16x4) over B tiles (4x16) holding the 256 per-batch diffs,
// accumulating through C. Fixed order -> bitwise deterministic.
// B-tile layout (f32, 4x16, 2 VGPRs): VGPR0 = rows k=0 (lanes 0-15) / k=2
// (lanes 16-31); VGPR1 = rows k=1 / k=3. Since all A rows are ones, every
// lane's D[...] VGPR0 holds colsum[n = lane & 15]; butterfly within the
// 16-lane group gives the total without double counting.
// =========================================================================
__global__ __launch_bounds__(32) void crf_reduce_kernel(
    const float* __restrict__ ws, float* __restrict__ out)
{
  const int l = threadIdx.x;
#if HAVE_WMMA_F32
  typedef float v2f __attribute__((ext_vector_type(2)));
  typedef float v8f __attribute__((ext_vector_type(8)));
  const int base = ((l >> 4) * 32) + (l & 15);  // k-row pair start for this lane
  v2f A;
  A[0] = 1.0f;
  A[1] = 1.0f;
  v8f acc = {};
#pragma unroll
  for (int c = 0; c < 4; ++c) {        // 4 chunks x 64 values = 256
    const int i0 = c * 64 + base;
    v2f Bv;
    Bv[0] = ws[i0]      - ws[Bn + i0];        // row k   (k=0 or 2)
    Bv[1] = ws[i0 + 16] - ws[Bn + i0 + 16];   // row k+1 (k=1 or 3)
    acc = __builtin_amdgcn_wmma_f32_16x16x4_f32(
        false, A, false, Bv, (short)0, acc, false, false);
  }
  float tot = acc[0];                  // colsum[l & 15] in every lane
#pragma unroll
  for (int off = 8; off; off >>= 1)    // reduce 16 distinct columns
    tot += __shfl_xor(tot, off, 32);
  if (l == 0) out[0] = tot;
#else
  // Fallback: 32-lane strided sum + full butterfly (deterministic).
  float v = 0.f;
#pragma unroll
  for (int c = 0; c < 8; ++c) {
    const int i0 = c * 32 + l;
    v += ws[i0] - ws[Bn + i0];
  }
#pragma unroll
  for (int off = 16; off; off >>= 1)
    v += __shfl_xor(v, off, 32);
  if (l == 0) out[0] = v;
#endif
}

extern "C" void kernel_launch(void* const* d_in, const int* in_sizes, int n_in,
                              void* d_out, int out_size, void* d_ws, size_t ws_size,
                              hipStream_t stream) {
  const float* em    = (const float*)d_in[0];   // [T,B,M] f32
  const int*   tags  = (const int*)d_in[1];     // [T,B] i32
  const float* wgt   = (const float*)d_in[2];   // [T,B] f32
  const int*   msk   = (const int*)d_in[3];     // [T,B] i32
  const float* trans = (const float*)d_in[4];   // [M,M] f32
  const float* st    = (const float*)d_in[5];   // [M]   f32
  const float* et    = (const float*)d_in[6];   // [M]   f32
  float* ws  = (float*)d_ws;                    // [0..255]=logZ, [256..511]=score
  float* out = (float*)d_out;

  crf_forward_kernel<<<Bn / 2, 32, 0, stream>>>(em, wgt, msk, trans, st, et, ws);
  crf_score_kernel<<<Bn, 256, 0, stream>>>(em, tags, wgt, msk, trans, st, et, ws + Bn);
  crf_reduce_kernel<<<1, 32, 0, stream>>>(ws, out);
}